// ClassBalancedLosses_55645596287217
// MI455X (gfx1250) — compile-verified
//
#include <hip/hip_runtime.h>
#include <hip/hip_bf16.h>

#define NCLASS 19
#define BHW    (8 * 512 * 1024)      // total pixels
#define HW     (512 * 1024)          // pixels per image
#define TP     256                   // tile pixels (tile_dim0)
#define TILES_PER_BLOCK 8
#define NTILES (BHW / TP)            // 16384
#define NBLK_MAIN (NTILES / TILES_PER_BLOCK)  // 2048

typedef unsigned int u32x4 __attribute__((ext_vector_type(4)));
typedef int          i32x8 __attribute__((ext_vector_type(8)));
typedef int          i32x4 __attribute__((ext_vector_type(4)));

// ---------------- kernel 1: zero counts ----------------
__global__ void cb_zero_counts(unsigned int* counts) {
    if (threadIdx.x < NCLASS) counts[threadIdx.x] = 0u;
}

// ---------------- kernel 2: global class histogram ----------------
__global__ __launch_bounds__(256) void cb_hist(const int* __restrict__ tgt,
                                               unsigned int* __restrict__ counts) {
    __shared__ unsigned int h[8][NCLASS + 1];   // per-wave sub-histograms (pad row)
    const int tid  = threadIdx.x;
    const int wave = tid >> 5;
    for (int i = tid; i < 8 * (NCLASS + 1); i += blockDim.x)
        ((unsigned int*)h)[i] = 0u;
    __syncthreads();

    const int4* t4 = (const int4*)tgt;
    const int n4 = BHW >> 2;
    for (int i = blockIdx.x * blockDim.x + tid; i < n4; i += gridDim.x * blockDim.x) {
        int4 v = t4[i];
        if ((unsigned)v.x < NCLASS) atomicAdd(&h[wave][v.x], 1u);
        if ((unsigned)v.y < NCLASS) atomicAdd(&h[wave][v.y], 1u);
        if ((unsigned)v.z < NCLASS) atomicAdd(&h[wave][v.z], 1u);
        if ((unsigned)v.w < NCLASS) atomicAdd(&h[wave][v.w], 1u);
    }
    __syncthreads();
    if (tid < NCLASS) {
        unsigned int s = 0;
        #pragma unroll
        for (int w = 0; w < 8; ++w) s += h[w][tid];
        if (s) atomicAdd(&counts[tid], s);  // integer atomics: deterministic
    }
}

// ---------------- kernel 3: class-balanced weights ----------------
__global__ void cb_weights(const unsigned int* __restrict__ counts,
                           float* __restrict__ weights) {
    int c = threadIdx.x;
    if (c < NCLASS) {
        unsigned int n = counts[c];
        float w = 0.f;
        if (n > 0) {
            double p = exp((double)n * log(0.999));   // beta^n
            w = (float)(0.001 / (1.0 - p));           // (1-beta)/(1-beta^n)
        }
        weights[c] = w;
    }
}

// ---------------- TDM tile load: 19 strided rows -> contiguous LDS ----------------
__device__ __forceinline__ void tdm_load_tile(const float* gtile, unsigned int lds_off) {
    unsigned long long ga = (unsigned long long)(uintptr_t)gtile;
    // D# group0: count=1, lds_addr, 57b global_addr, type=2 (ISA 8.3)
    u32x4 g0 = { 1u,
                 lds_off,
                 (unsigned int)ga,
                 (unsigned int)((ga >> 32) & 0x1FFFFFFu) | 0x80000000u };
    // D# group1 (ISA 8.4): data_size=4B; tensor_dim0=HW, tensor_dim1=19;
    // tile_dim0=TP, tile_dim1=19, tile_dim2=0; dim0_stride=HW; no pad/iterate/mask
    i32x8 g1 = { 0x00020000,                      // data_size=2 (4 bytes)
                 (int)((HW & 0xFFFF) << 16),      // tensor_dim0[15:0]
                 (int)((HW >> 16) | (NCLASS << 16)),  // tensor_dim0[31:16] | tensor_dim1[15:0]
                 (int)(TP << 16),                 // tensor_dim1[31:16]=0 | tile_dim0
                 (int)NCLASS,                     // tile_dim1 | tile_dim2=0
                 (int)HW,                         // tensor_dim0_stride[31:0]
                 0, 0 };                          // stride0[47:32], stride1 (unused 2D)
    i32x4 z4 = { 0, 0, 0, 0 };
    i32x8 z8 = { 0, 0, 0, 0, 0, 0, 0, 0 };
    // 6-arg form (clang-23 / therock-10.0): extra int32x8 group before cpol
    __builtin_amdgcn_tensor_load_to_lds(g0, g1, z4, z4, z8, 0);
}

// ---------------- kernel 4: main streaming loss ----------------
__global__ __launch_bounds__(256) void cb_main(const float* __restrict__ logits,
                                               const int* __restrict__ target,
                                               const float* __restrict__ weights,
                                               float* __restrict__ partials) {
    __shared__ float s_buf[2][NCLASS * TP];   // double-buffered 19x256 tiles (38 KB)
    __shared__ float s_w[32];
    __shared__ float s_red[512];

    const int tid  = threadIdx.x;
    const int wave = tid >> 5;
    if (tid < NCLASS) s_w[tid] = weights[tid];

    const int tile0 = blockIdx.x * TILES_PER_BLOCK;

    if (wave == 0) {
        int t = tile0;
        const float* g = logits + ((unsigned long long)(t >> 11) * NCLASS * HW
                                   + (unsigned long long)((t & 2047) << 8));
        tdm_load_tile(g, (unsigned int)(uintptr_t)&s_buf[0][0]);
    }

    float acc_wl = 0.f, acc_w = 0.f;

    #pragma unroll
    for (int i = 0; i < TILES_PER_BLOCK; ++i) {
        const int tile = tile0 + i;
        if (wave == 0) {
            if (i + 1 < TILES_PER_BLOCK) {
                int t = tile + 1;
                const float* g = logits + ((unsigned long long)(t >> 11) * NCLASS * HW
                                           + (unsigned long long)((t & 2047) << 8));
                tdm_load_tile(g, (unsigned int)(uintptr_t)&s_buf[(i + 1) & 1][0]);
                __builtin_amdgcn_s_wait_tensorcnt(1);  // tile i landed (in-order)
            } else {
                __builtin_amdgcn_s_wait_tensorcnt(0);
            }
        }
        __syncthreads();                                // tile i visible to all waves

        const float* __restrict__ L = &s_buf[i & 1][0];
        const int b  = tile >> 11;
        const int p0 = (tile & 2047) << 8;
        const int p  = tid;                             // TP==blockDim.x

        float m = L[p];
        #pragma unroll
        for (int c = 1; c < NCLASS; ++c) m = fmaxf(m, L[c * TP + p]);
        float s = 0.f;
        #pragma unroll
        for (int c = 0; c < NCLASS; ++c) s += __expf(L[c * TP + p] - m);
        const float lse = m + __logf(s);

        const int t  = target[(unsigned long long)b * HW + p0 + p];
        const int tc = min(max(t, 0), NCLASS - 1);
        const float w = (t != -1) ? s_w[tc] : 0.f;
        acc_wl += w * (L[tc * TP + p] - lse);
        acc_w  += w;
        __syncthreads();                                // done reading buf[i&1]
    }

    // deterministic block tree-reduction
    s_red[tid] = acc_wl;
    s_red[256 + tid] = acc_w;
    __syncthreads();
    #pragma unroll
    for (int off = 128; off > 0; off >>= 1) {
        if (tid < off) {
            s_red[tid]       += s_red[tid + off];
            s_red[256 + tid] += s_red[256 + tid + off];
        }
        __syncthreads();
    }
    if (tid == 0) {
        partials[2 * blockIdx.x]     = s_red[0];
        partials[2 * blockIdx.x + 1] = s_red[256];
    }
}

// ---------------- kernel 5: finalize ----------------
__global__ __launch_bounds__(256) void cb_finalize(const float* __restrict__ partials,
                                                   float* __restrict__ out) {
    __shared__ float r[512];
    const int tid = threadIdx.x;
    float wl = 0.f, w = 0.f;
    for (int i = tid; i < NBLK_MAIN; i += 256) {        // fixed order per thread
        wl += partials[2 * i];
        w  += partials[2 * i + 1];
    }
    r[tid] = wl; r[256 + tid] = w;
    __syncthreads();
    #pragma unroll
    for (int off = 128; off > 0; off >>= 1) {
        if (tid < off) { r[tid] += r[tid + off]; r[256 + tid] += r[256 + tid + off]; }
        __syncthreads();
    }
    if (tid == 0) out[0] = -r[0] / r[256];
}

extern "C" void kernel_launch(void* const* d_in, const int* in_sizes, int n_in,
                              void* d_out, int out_size, void* d_ws, size_t ws_size,
                              hipStream_t stream) {
    const float* logits = (const float*)d_in[0];
    const int*   target = (const int*)d_in[1];
    float* out = (float*)d_out;

    char* ws = (char*)d_ws;
    unsigned int* counts  = (unsigned int*)ws;          // 19 u32
    float*        weights = (float*)(ws + 128);         // 19 f32
    float*        partials= (float*)(ws + 256);         // 2*NBLK_MAIN f32

    cb_zero_counts<<<1, 32, 0, stream>>>(counts);
    cb_hist<<<256, 256, 0, stream>>>(target, counts);
    cb_weights<<<1, 32, 0, stream>>>(counts, weights);
    cb_main<<<NBLK_MAIN, 256, 0, stream>>>(logits, target, weights, partials);
    cb_finalize<<<1, 256, 0, stream>>>(partials, out);
}